// GATDefectDetectionModel_23261542875617
// MI455X (gfx1250) — compile-verified
//
#include <hip/hip_runtime.h>
#include <hip/hip_bf16.h>

// ---------------------------------------------------------------------------
// GAT defect-detection model for MI455X (gfx1250, wave32, WMMA + async-LDS).
// Per layer: h = X@W (fp32 WMMA 16x16x4; B panels DMA'd to LDS with
// GLOBAL_LOAD_ASYNC_TO_LDS_B128) -> s_src/s_dst row dots -> edge segment-max
// (encoded atomicMax) -> fused exp/denom/feature scatter (atomic f32 adds,
// L2-resident) -> /denom + bias -> batchnorm -> relu. Then mean-pool + MLP.
// ---------------------------------------------------------------------------

typedef __attribute__((ext_vector_type(2))) float v2f;
typedef __attribute__((ext_vector_type(8))) float v8f;
typedef int v4i __attribute__((vector_size(16)));
typedef __attribute__((address_space(1))) v4i as1_v4i;   // global
typedef __attribute__((address_space(3))) v4i as3_v4i;   // LDS

#define GAT_F 256          // feature width of both GAT layers
#define KP 64              // K panel staged in LDS per iteration
#define LDS_STRIDE (KP + 4)  // +4 floats pad: ds_load_b64 bank spacing of 4

#if __has_builtin(__builtin_amdgcn_global_load_async_to_lds_b128)
#define HAVE_ASYNC_LDS 1
#else
#define HAVE_ASYNC_LDS 0
#endif

// ---------------- weight transpose: Wt[N,K] = W[K,N]^T ---------------------
__global__ __launch_bounds__(256) void transpose_w(
    const float* __restrict__ W, float* __restrict__ Wt, int K, int Ncols) {
  const int idx = blockIdx.x * 256 + threadIdx.x;
  if (idx >= K * Ncols) return;
  const int k = idx / Ncols;
  const int n = idx - k * Ncols;
  Wt[(size_t)n * K + k] = W[idx];
}

// ---------------- fp32 WMMA GEMM: C[M,N] = A[M,K] @ Bt[N,K]^T --------------
// Block (8 waves) computes a 32x256 tile; each wave a 16x64 strip (4 accs).
// K is consumed in KP=64 panels: the 256xKP B panel is async-DMA'd into LDS
// (64KB + pad), then each wave runs 16 k-steps x 4 WMMAs off ds_load_b64.
// Requires M % 32 == 0, K % KP == 0, Ncols == 256 (true for this model).
// A 16x4 layout : lanes 0-15 -> M=lane, VGPR0=K0,VGPR1=K1; lanes 16-31 -> K2,K3
// B 4x16 layout : lanes 0-15 -> N=lane, VGPR0=K0,VGPR1=K1; lanes 16-31 -> K2,K3
// C/D layout    : VGPR v, lanes 0-15 -> (M=v, N=lane); lanes 16-31 -> (M=v+8)
__global__ __launch_bounds__(256) void gemm_wmma_f32_bt(
    const float* __restrict__ A, const float* __restrict__ Bt,
    float* __restrict__ C, int M, int K, int Ncols) {
  __shared__ float bsh[GAT_F * LDS_STRIDE];  // 256*68*4 = 69,632 B

  const int t = threadIdx.x;
  const int lane = t & 31;
  const int wave = t >> 5;           // 0..7
  const int mw = wave >> 2;          // 0..1 -> 16-row half of the 32-row tile
  const int nw = wave & 3;           // 0..3 -> 64-col strip
  const int m0 = blockIdx.x * 32 + mw * 16;
  const int n0 = nw * 64;
  const int rc = lane & 15;          // row for A, col for B/C
  const int kk = (lane >> 4) << 1;   // 0 or 2

  v8f c0 = {}, c1 = {}, c2 = {}, c3 = {};
  const float* ap = A + (size_t)(m0 + rc) * K + kk;

  for (int kp = 0; kp < K; kp += KP) {
    // ---- stage B panel (256 cols x KP) into LDS ----
    // panel = 256*KP/4 = 4096 float4s; 16 per thread.
#pragma unroll
    for (int i = 0; i < (GAT_F * KP / 4) / 256; ++i) {
      const int idx = t + i * 256;          // float4 index within panel
      const int col = idx >> 4;             // KP/4 = 16 float4 per column
      const int kq = idx & 15;
      const float* g = Bt + (size_t)col * K + kp + kq * 4;
      float* l = &bsh[col * LDS_STRIDE + kq * 4];
#if HAVE_ASYNC_LDS
      __builtin_amdgcn_global_load_async_to_lds_b128(
          (as1_v4i*)g, (as3_v4i*)l, 0, 0);
#else
      *(float4*)l = *(const float4*)g;
#endif
    }
#if HAVE_ASYNC_LDS
#if __has_builtin(__builtin_amdgcn_s_wait_asynccnt)
    __builtin_amdgcn_s_wait_asynccnt(0);
#else
    asm volatile("s_wait_asynccnt 0x0" ::: "memory");
#endif
#endif
    __syncthreads();

    // ---- compute on the panel ----
    const float* bl0 = &bsh[(n0 + 0 + rc) * LDS_STRIDE + kk];
    const float* bl1 = &bsh[(n0 + 16 + rc) * LDS_STRIDE + kk];
    const float* bl2 = &bsh[(n0 + 32 + rc) * LDS_STRIDE + kk];
    const float* bl3 = &bsh[(n0 + 48 + rc) * LDS_STRIDE + kk];
    const float* apk = ap + kp;
#pragma unroll
    for (int k = 0; k < KP; k += 4) {
      v2f a  = *(const v2f*)(apk + k);
      v2f b0 = *(const v2f*)(bl0 + k);
      v2f b1 = *(const v2f*)(bl1 + k);
      v2f b2 = *(const v2f*)(bl2 + k);
      v2f b3 = *(const v2f*)(bl3 + k);
      c0 = __builtin_amdgcn_wmma_f32_16x16x4_f32(false, a, false, b0, (short)0,
                                                 c0, false, false);
      c1 = __builtin_amdgcn_wmma_f32_16x16x4_f32(false, a, false, b1, (short)0,
                                                 c1, false, false);
      c2 = __builtin_amdgcn_wmma_f32_16x16x4_f32(false, a, false, b2, (short)0,
                                                 c2, false, false);
      c3 = __builtin_amdgcn_wmma_f32_16x16x4_f32(false, a, false, b3, (short)0,
                                                 c3, false, false);
    }
    __syncthreads();
  }

  const int rb = (lane >> 4) << 3;   // 0 or 8
  float* cp = C + (size_t)(m0 + rb) * Ncols + rc;
#pragma unroll
  for (int v = 0; v < 8; ++v) {
    float* r = cp + (size_t)v * Ncols + n0;
    r[0]  = c0[v];
    r[16] = c1[v];
    r[32] = c2[v];
    r[48] = c3[v];
  }
}

// ------------- per-node dot products: s_src = h.a_s, s_dst = h.a_d ---------
__global__ __launch_bounds__(256) void row_dots(
    const float* __restrict__ H, const float* __restrict__ a_s,
    const float* __restrict__ a_d, float* __restrict__ s_src,
    float* __restrict__ s_dst, int Nn) {
  const int lane = threadIdx.x & 31;
  const int node = blockIdx.x * 8 + (threadIdx.x >> 5);
  if (node >= Nn) return;
  const float* hp = H + (size_t)node * GAT_F;
  float s1 = 0.f, s2 = 0.f;
  for (int f = lane; f < GAT_F; f += 32) {
    float v = hp[f];
    s1 += v * a_s[f];
    s2 += v * a_d[f];
  }
  for (int o = 16; o > 0; o >>= 1) {
    s1 += __shfl_xor(s1, o, 32);
    s2 += __shfl_xor(s2, o, 32);
  }
  if (lane == 0) { s_src[node] = s1; s_dst[node] = s2; }
}

// ------------- order-preserving float<->uint encode for atomic max ---------
__device__ __forceinline__ unsigned enc_f(float f) {
  unsigned u = __float_as_uint(f);
  return (u & 0x80000000u) ? ~u : (u | 0x80000000u);
}
__device__ __forceinline__ float dec_f(unsigned u) {
  return (u & 0x80000000u) ? __uint_as_float(u ^ 0x80000000u)
                           : __uint_as_float(~u);
}

__device__ __forceinline__ void edge_endpoints(const int* __restrict__ ei,
                                               int E, int i, int& s, int& d) {
  if (i < E) { s = ei[i]; d = ei[E + i]; }  // edge_index row-major (2,E)
  else       { s = d = i - E; }             // appended self loops
}

// ------------- pass 1: segment max of leaky_relu attention logits ----------
__global__ void edge_max(const int* __restrict__ ei, int E, int EE,
                         const float* __restrict__ s_src,
                         const float* __restrict__ s_dst,
                         unsigned* __restrict__ m_enc) {
  const int i = blockIdx.x * blockDim.x + threadIdx.x;
  if (i >= EE) return;
  int s, d;
  edge_endpoints(ei, E, i, s, d);
  float e = s_src[s] + s_dst[d];
  e = (e > 0.f) ? e : 0.2f * e;
  atomicMax(&m_enc[d], enc_f(e));
}

// ------------- pass 2: fused exp, denom accumulation, feature scatter ------
// One wave per edge; 256-float message row gathered with b128 loads from
// h[src] and atomic-added into acc[dst] (both L2-resident on the 192MB L2).
__global__ __launch_bounds__(256) void edge_scatter(
    const int* __restrict__ ei, int E, int EE,
    const float* __restrict__ s_src, const float* __restrict__ s_dst,
    const unsigned* __restrict__ m_enc, float* __restrict__ denom,
    const float* __restrict__ H, float* __restrict__ acc) {
  const int lane = threadIdx.x & 31;
  const int i = blockIdx.x * 8 + (threadIdx.x >> 5);
  if (i >= EE) return;
  int s, d;
  edge_endpoints(ei, E, i, s, d);
  float e = s_src[s] + s_dst[d];
  e = (e > 0.f) ? e : 0.2f * e;
  const float ex = __expf(e - dec_f(m_enc[d]));
  if (lane == 0) atomicAdd(&denom[d], ex);
  const float4* hs = (const float4*)(H + (size_t)s * GAT_F);
  float* ad = acc + (size_t)d * GAT_F;
#pragma unroll
  for (int f4 = lane; f4 < GAT_F / 4; f4 += 32) {
    const float4 v = hs[f4];
    const int f = f4 * 4;
    atomicAdd(&ad[f + 0], ex * v.x);
    atomicAdd(&ad[f + 1], ex * v.y);
    atomicAdd(&ad[f + 2], ex * v.z);
    atomicAdd(&ad[f + 3], ex * v.w);
  }
}

// ------------- out = acc/denom + bias (in place) ---------------------------
__global__ __launch_bounds__(256) void node_finalize(
    float* __restrict__ acc, const float* __restrict__ denom,
    const float* __restrict__ bias) {
  const size_t idx = (size_t)blockIdx.x * GAT_F + threadIdx.x;
  acc[idx] = acc[idx] / denom[blockIdx.x] + bias[threadIdx.x];
}

// ------------- batchnorm column statistics ---------------------------------
#define STAT_ROWS 256
__global__ __launch_bounds__(256) void col_stats(
    const float* __restrict__ X, int Nn, float* __restrict__ sum,
    float* __restrict__ sumsq) {
  const int f = threadIdx.x;
  const int r0 = blockIdx.x * STAT_ROWS;
  const int r1 = min(r0 + STAT_ROWS, Nn);
  float s = 0.f, q = 0.f;
  for (int r = r0; r < r1; ++r) {
    float v = X[(size_t)r * GAT_F + f];
    s += v;
    q += v * v;
  }
  atomicAdd(&sum[f], s);
  atomicAdd(&sumsq[f], q);
}

__global__ void stats_finalize(const float* __restrict__ sum,
                               const float* __restrict__ sumsq,
                               const float* __restrict__ g,
                               const float* __restrict__ be,
                               float* __restrict__ scale,
                               float* __restrict__ shift, int Nn) {
  const int f = threadIdx.x;
  const float inv_n = 1.0f / (float)Nn;
  const float mu = sum[f] * inv_n;
  const float var = sumsq[f] * inv_n - mu * mu;
  const float sc = rsqrtf(var + 1e-5f) * g[f];
  scale[f] = sc;
  shift[f] = be[f] - mu * sc;
}

__global__ __launch_bounds__(256) void bn_relu(
    const float* __restrict__ X, const float* __restrict__ scale,
    const float* __restrict__ shift, float* __restrict__ Y) {
  const size_t idx = (size_t)blockIdx.x * GAT_F + threadIdx.x;
  const float v = X[idx] * scale[threadIdx.x] + shift[threadIdx.x];
  Y[idx] = v > 0.f ? v : 0.f;
}

// ------------- global mean pool (sums + counts) ----------------------------
__global__ __launch_bounds__(256) void pool_sum(
    const float* __restrict__ X, const int* __restrict__ batch,
    float* __restrict__ pooled, float* __restrict__ cnt) {
  const int n = blockIdx.x;
  const int f = threadIdx.x;
  const int g = batch[n];
  atomicAdd(&pooled[(size_t)g * GAT_F + f], X[(size_t)n * GAT_F + f]);
  if (f == 0) atomicAdd(&cnt[g], 1.0f);
}

// ------------- tiny MLP head ----------------------------------------------
__global__ __launch_bounds__(256) void mlp_fc1(
    const float* __restrict__ pooled, const float* __restrict__ cnt,
    const float* __restrict__ W, const float* __restrict__ b,
    float* __restrict__ Z) {
  const int g = blockIdx.x, j = threadIdx.x;  // 256 outputs
  const float ic = 1.0f / fmaxf(cnt[g], 1.0f);
  const float* p = pooled + (size_t)g * GAT_F;
  float s = b[j];
  for (int k = 0; k < GAT_F; ++k) s += (p[k] * ic) * W[(size_t)k * 256 + j];
  Z[(size_t)g * 256 + j] = fmaxf(s, 0.f);
}

__global__ __launch_bounds__(128) void mlp_fc2(
    const float* __restrict__ Z1, const float* __restrict__ W,
    const float* __restrict__ b, float* __restrict__ Z2) {
  const int g = blockIdx.x, j = threadIdx.x;  // 128 outputs
  const float* p = Z1 + (size_t)g * 256;
  float s = b[j];
  for (int k = 0; k < 256; ++k) s += p[k] * W[(size_t)k * 128 + j];
  Z2[(size_t)g * 128 + j] = fmaxf(s, 0.f);
}

__global__ void mlp_fc3(const float* __restrict__ Z2,
                        const float* __restrict__ W,
                        const float* __restrict__ b, float* __restrict__ out) {
  const int g = threadIdx.x;  // one thread per graph
  const float* p = Z2 + (size_t)g * 128;
  float s = b[0];
  for (int k = 0; k < 128; ++k) s += p[k] * W[k];
  out[g] = s;
}

// ---------------------------------------------------------------------------
extern "C" void kernel_launch(void* const* d_in, const int* in_sizes, int n_in,
                              void* d_out, int out_size, void* d_ws,
                              size_t ws_size, hipStream_t stream) {
  const float* x      = (const float*)d_in[0];
  const float* W1     = (const float*)d_in[1];
  const float* a_src1 = (const float*)d_in[2];
  const float* a_dst1 = (const float*)d_in[3];
  const float* b1     = (const float*)d_in[4];
  const float* g1     = (const float*)d_in[5];
  const float* be1    = (const float*)d_in[6];
  const float* W2     = (const float*)d_in[7];
  const float* a_src2 = (const float*)d_in[8];
  const float* a_dst2 = (const float*)d_in[9];
  const float* b2     = (const float*)d_in[10];
  const float* g2     = (const float*)d_in[11];
  const float* be2    = (const float*)d_in[12];
  const float* fc1_w  = (const float*)d_in[13];
  const float* fc1_b  = (const float*)d_in[14];
  const float* fc2_w  = (const float*)d_in[15];
  const float* fc2_b  = (const float*)d_in[16];
  const float* fc3_w  = (const float*)d_in[17];
  const float* fc3_b  = (const float*)d_in[18];
  const int*   ei     = (const int*)d_in[19];
  const int*   batch  = (const int*)d_in[20];
  float* out = (float*)d_out;

  const int IN = 128;
  const int G  = 64;
  const int Nn = in_sizes[0] / IN;   // 100000 (multiple of 32)
  const int E  = in_sizes[19] / 2;   // 3200000
  const int EE = E + Nn;             // edges + self loops

  // ---- workspace layout (floats) ----
  float* ws = (float*)d_ws;
  const size_t NH = (size_t)Nn * GAT_F;
  float*    hbuf   = ws;                         // [Nn,256] linear output
  float*    act    = hbuf + NH;                  // [Nn,256] layer activation
  float*    acc    = act + NH;                   // [Nn,256] scatter accumulator
  float*    s_src  = acc + NH;                   // [Nn]
  float*    s_dst  = s_src + Nn;                 // [Nn]
  float*    denom  = s_dst + Nn;                 // [Nn]
  unsigned* m_enc  = (unsigned*)(denom + Nn);    // [Nn]
  float*    colsum = (float*)(m_enc + Nn);       // [256]
  float*    colsq  = colsum + 256;               // [256]
  float*    scale  = colsq + 256;                // [256]
  float*    shift  = scale + 256;                // [256]
  float*    pooled = shift + 256;                // [G,256]
  float*    cnt    = pooled + (size_t)G * 256;   // [G]
  float*    z1     = cnt + G;                    // [G,256]
  float*    z2     = z1 + (size_t)G * 256;       // [G,128]
  float*    wt1    = z2 + (size_t)G * 128;       // [256,IN]  W1^T
  float*    wt2    = wt1 + (size_t)GAT_F * IN;   // [256,256] W2^T
  float*    ws_end = wt2 + (size_t)GAT_F * GAT_F;
  (void)ws_end; (void)ws_size; (void)n_in; (void)out_size;

  // region zeroed once per layer: acc..colsq (acc, s_src, s_dst, denom,
  // m_enc [0 == encoded -inf], colsum, colsq)
  const size_t zeroA = (size_t)((char*)(colsq + 256) - (char*)acc);
  const size_t zeroP = (size_t)((char*)(z2 + (size_t)G * 128) - (char*)pooled);

  const int  gemmBlocks = Nn / 32;               // 3125 blocks of 8 waves
  const int  nodeWaveBlocks = (Nn + 7) / 8;
  const int  edgeThreadBlocks = (EE + 255) / 256;
  const int  edgeWaveBlocks = (EE + 7) / 8;
  const int  statBlocks = (Nn + STAT_ROWS - 1) / STAT_ROWS;

  // ---- weight transposes (weights stay L2-resident) ----
  transpose_w<<<(IN * GAT_F + 255) / 256, 256, 0, stream>>>(W1, wt1, IN,
                                                            GAT_F);
  transpose_w<<<(GAT_F * GAT_F + 255) / 256, 256, 0, stream>>>(W2, wt2, GAT_F,
                                                               GAT_F);

  // ================= layer 1 =================
  (void)hipMemsetAsync(acc, 0, zeroA, stream);
  gemm_wmma_f32_bt<<<gemmBlocks, 256, 0, stream>>>(x, wt1, hbuf, Nn, IN,
                                                   GAT_F);
  row_dots<<<nodeWaveBlocks, 256, 0, stream>>>(hbuf, a_src1, a_dst1, s_src,
                                               s_dst, Nn);
  edge_max<<<edgeThreadBlocks, 256, 0, stream>>>(ei, E, EE, s_src, s_dst,
                                                 m_enc);
  edge_scatter<<<edgeWaveBlocks, 256, 0, stream>>>(ei, E, EE, s_src, s_dst,
                                                   m_enc, denom, hbuf, acc);
  node_finalize<<<Nn, 256, 0, stream>>>(acc, denom, b1);
  col_stats<<<statBlocks, 256, 0, stream>>>(acc, Nn, colsum, colsq);
  stats_finalize<<<1, 256, 0, stream>>>(colsum, colsq, g1, be1, scale, shift,
                                        Nn);
  bn_relu<<<Nn, 256, 0, stream>>>(acc, scale, shift, act);

  // ================= layer 2 =================
  (void)hipMemsetAsync(acc, 0, zeroA, stream);
  gemm_wmma_f32_bt<<<gemmBlocks, 256, 0, stream>>>(act, wt2, hbuf, Nn, GAT_F,
                                                   GAT_F);
  row_dots<<<nodeWaveBlocks, 256, 0, stream>>>(hbuf, a_src2, a_dst2, s_src,
                                               s_dst, Nn);
  edge_max<<<edgeThreadBlocks, 256, 0, stream>>>(ei, E, EE, s_src, s_dst,
                                                 m_enc);
  edge_scatter<<<edgeWaveBlocks, 256, 0, stream>>>(ei, E, EE, s_src, s_dst,
                                                   m_enc, denom, hbuf, acc);
  node_finalize<<<Nn, 256, 0, stream>>>(acc, denom, b2);
  col_stats<<<statBlocks, 256, 0, stream>>>(acc, Nn, colsum, colsq);
  stats_finalize<<<1, 256, 0, stream>>>(colsum, colsq, g2, be2, scale, shift,
                                        Nn);
  bn_relu<<<Nn, 256, 0, stream>>>(acc, scale, shift, act);

  // ================= pool + MLP =================
  (void)hipMemsetAsync(pooled, 0, zeroP, stream);
  pool_sum<<<Nn, 256, 0, stream>>>(act, batch, pooled, cnt);
  mlp_fc1<<<G, 256, 0, stream>>>(pooled, cnt, fc1_w, fc1_b, z1);
  mlp_fc2<<<G, 128, 0, stream>>>(z1, fc2_w, fc2_b, z2);
  mlp_fc3<<<1, G, 0, stream>>>(z2, fc3_w, fc3_b, out);
}